// ProtoryNet_87625922773156
// MI455X (gfx1250) — compile-verified
//
#include <hip/hip_runtime.h>
#include <hip/hip_bf16.h>

// ---------------------------------------------------------------------------
// ProtoryNet distance layer for MI455X (gfx1250, wave32).
// distances = ||x||^2 + ||p||^2 - 2 x.p   via V_WMMA_F32_16X16X4_F32
// dist_hot  = softmax(-1e6 * d) * (exp(-0.1 d) + 1e-8)
// ---------------------------------------------------------------------------

typedef float v2f __attribute__((ext_vector_type(2)));
typedef float v8f __attribute__((ext_vector_type(8)));

#define BDIM      16
#define NDIM      256
#define DDIM      512
#define KDIM      20
#define BN        (BDIM * NDIM)      // 4096 rows
#define HOT_ELEMS (BN * KDIM)        // 81920
#define PSTRIDE   516                // 516 % 64 == 4 -> conflict-free B frags
#define A_DECAY   0.1f
#define BETA_SM   1.0e6f
#define EPS_SM    1e-8f

__device__ __forceinline__ float half16_max(float v) {
    // reduce across lanes 0-15 (and independently 16-31): xor masks stay in-half
    v = fmaxf(v, __shfl_xor(v, 1, 32));
    v = fmaxf(v, __shfl_xor(v, 2, 32));
    v = fmaxf(v, __shfl_xor(v, 4, 32));
    v = fmaxf(v, __shfl_xor(v, 8, 32));
    return v;
}
__device__ __forceinline__ float half16_sum(float v) {
    v += __shfl_xor(v, 1, 32);
    v += __shfl_xor(v, 2, 32);
    v += __shfl_xor(v, 4, 32);
    v += __shfl_xor(v, 8, 32);
    return v;
}

__global__ __launch_bounds__(256)
void protorynet_dist_wmma(const float* __restrict__ x,
                          const float* __restrict__ protos,
                          float* __restrict__ out) {
    // LDS: 21 prototype rows (row 20 = zeros for the padded col tile) + norms
    __shared__ float sP[21 * PSTRIDE];   // ~43.3 KB
    __shared__ float sp2[32];

    const int tid = threadIdx.x;

    // ---- stage prototypes (rows 0..19 real, row 20 zero), coalesced ----
    for (int idx = tid; idx < 21 * DDIM; idx += 256) {
        const int n = idx >> 9;          // / 512
        const int c = idx & (DDIM - 1);
        sP[n * PSTRIDE + c] = (n < KDIM) ? protos[(n << 9) + c] : 0.0f;
    }
    // ---- prototype squared norms ----
    if (tid < 21) {
        float s = 0.0f;
        if (tid < KDIM) {
            const float* pr = protos + (tid << 9);
            for (int c = 0; c < DDIM; ++c) s = __builtin_fmaf(pr[c], pr[c], s);
        }
        sp2[tid] = s;
    }
    __syncthreads();

    const int lane    = tid & 31;
    const int wave    = tid >> 5;
    const int mloc    = lane & 15;       // tile-local row (A) / column (B,C)
    const int hi      = lane >> 4;       // half-wave select
    const int rowbase = (blockIdx.x * 8 + wave) * 16;

    // A fragment stream: lane covers row (rowbase+mloc), K = d0 + 2*hi + {0,1}
    const v2f* aV = (const v2f*)(x + (size_t)(rowbase + mloc) * DDIM + (hi << 1));
    // B fragment streams from LDS (proto row-major => B[k][n] = P[n][k])
    const int p1 = (16 + mloc < KDIM) ? (16 + mloc) : KDIM;  // clamp to zero row
    const v2f* b0V = (const v2f*)(sP + mloc * PSTRIDE + (hi << 1));
    const v2f* b1V = (const v2f*)(sP + p1   * PSTRIDE + (hi << 1));

    v8f acc0 = {};   // xp for protos 0..15
    v8f acc1 = {};   // xp for protos 16..31 (20..31 are zero)
    float ssq = 0.0f;

#pragma unroll 4
    for (int d2 = 0; d2 < DDIM / 2; d2 += 2) {   // d2 = d0/2, step K=4
        const v2f a  = aV[d2];
        const v2f b0 = b0V[d2];
        const v2f b1 = b1V[d2];
        ssq = __builtin_fmaf(a.x, a.x, ssq);
        ssq = __builtin_fmaf(a.y, a.y, ssq);
        acc0 = __builtin_amdgcn_wmma_f32_16x16x4_f32(
                   false, a, false, b0, (short)0, acc0, false, false);
        acc1 = __builtin_amdgcn_wmma_f32_16x16x4_f32(
                   false, a, false, b1, (short)0, acc1, false, false);
    }

    // full ||x_row||^2: lanes M and M+16 hold disjoint K halves of row M
    const float x2full = ssq + __shfl_xor(ssq, 16, 32);

    const float p2a = sp2[mloc];                      // proto mloc
    const int   n1  = 16 + mloc;
    const bool  v1  = (mloc < KDIM - 16);             // mloc < 4
    const float p2b = sp2[p1];                        // zero row if invalid

    float* __restrict__ hot = out;                    // [4096,20]
    float* __restrict__ dst = out + HOT_ELEMS;        // [4096,20]

#pragma unroll
    for (int i = 0; i < 8; ++i) {
        // row handled by this (i, half): rowbase + i + 8*hi
        const float x2r = __shfl(x2full, i + (hi << 3), 32);

        const float d0v = x2r + p2a - 2.0f * acc0[i];
        const float d1v = x2r + p2b - 2.0f * acc1[i];

        // stable softmax(-beta*d) across k = 0..19 (cols across the half-wave)
        const float ns0 = -BETA_SM * d0v;
        const float ns1 = v1 ? (-BETA_SM * d1v) : -__builtin_inff();
        const float mx  = half16_max(fmaxf(ns0, ns1));
        const float e0  = __expf(ns0 - mx);
        const float e1  = v1 ? __expf(ns1 - mx) : 0.0f;
        const float inv = 1.0f / half16_sum(e0 + e1);

        const float ed0 = __expf(-A_DECAY * d0v) + EPS_SM;
        const float ed1 = __expf(-A_DECAY * d1v) + EPS_SM;

        const int    r = rowbase + i + (hi << 3);
        const size_t o = (size_t)r * KDIM + mloc;
        hot[o] = (e0 * inv) * ed0;
        dst[o] = d0v;
        if (v1) {
            hot[o + 16] = (e1 * inv) * ed1;
            dst[o + 16] = d1v;
        }
    }
}

extern "C" void kernel_launch(void* const* d_in, const int* in_sizes, int n_in,
                              void* d_out, int out_size, void* d_ws, size_t ws_size,
                              hipStream_t stream) {
    (void)in_sizes; (void)n_in; (void)out_size; (void)d_ws; (void)ws_size;
    const float* x  = (const float*)d_in[0];   // [16,256,512]
    const float* pr = (const float*)d_in[1];   // [20,512]
    float* out = (float*)d_out;                // hot[4096,20] ++ dist[4096,20]

    dim3 grid(BN / 128);    // 32 blocks, 8 waves each, 16 rows per wave
    dim3 block(256);
    protorynet_dist_wmma<<<grid, block, 0, stream>>>(x, pr, out);
}